// GATWithJK_82437602280149
// MI455X (gfx1250) — compile-verified
//
#include <hip/hip_runtime.h>

// GATWithJK on MI455X (gfx1250, wave32).
// N=30000 nodes, E=480000 edges (+N self loops), D=128, H=4 (layers 0,1), H=1 (layer 2).
// Memory-bound workload (~1.6 GB edge gather/scatter traffic -> ~70us @ 23.3 TB/s);
// dense projections use V_WMMA_F32_16X16X4_F32 (exact f32) with W staged in LDS.

#define NEG_SLOPE 0.2f
#define LN_EPS 1e-5f

typedef float v2f __attribute__((ext_vector_type(2)));
typedef float v8f __attribute__((ext_vector_type(8)));

// ---------------------------------------------------------------------------
// hp[n,128] = h[n,128] @ W[128,128], f32 WMMA 16x16x4.
// Block = 256 threads = 8 waves. Block b handles M-tile b (rows 16b..16b+15);
// wave w handles N-tile w (cols 16w..16w+15). W (64 KB) staged in LDS.
// ---------------------------------------------------------------------------
__global__ __launch_bounds__(256) void gemm128_wmma(
    const float* __restrict__ h, const float* __restrict__ W,
    float* __restrict__ hp)
{
    __shared__ float Ws[128 * 128];           // 64 KB LDS
    {
        const float4* Wv  = (const float4*)W;
        float4*       Wsv = (float4*)Ws;
#pragma unroll
        for (int i = 0; i < (128 * 128 / 4) / 256; ++i)
            Wsv[i * 256 + threadIdx.x] = Wv[i * 256 + threadIdx.x];
    }
    __syncthreads();

    const int lane  = threadIdx.x & 31;
    const int wave  = threadIdx.x >> 5;       // N-tile 0..7
    const int row   = lane & 15;              // M row within tile (A); N col (B, store)
    const int khalf = (lane >> 4) << 1;       // 0 or 2: K pair this half-wave owns
    const int tileM = blockIdx.x;

    const float* arow = h + (size_t)(tileM * 16 + row) * 128;
    const int    col  = wave * 16 + row;

    v8f c = {};
#pragma unroll
    for (int k0 = 0; k0 < 128; k0 += 4) {
        v2f a, b;
        // A (16x4): VGPR0 = K=khalf, VGPR1 = K=khalf+1 per ISA layout
        a.x = arow[k0 + khalf];
        a.y = arow[k0 + khalf + 1];
        // B (4x16): row K striped across lanes, same half-wave K split
        b.x = Ws[(k0 + khalf) * 128 + col];
        b.y = Ws[(k0 + khalf + 1) * 128 + col];
        c = __builtin_amdgcn_wmma_f32_16x16x4_f32(
                /*neg_a=*/false, a, /*neg_b=*/false, b,
                /*c_mod=*/(short)0, c, /*reuse_a=*/false, /*reuse_b=*/false);
    }

    // C/D layout: VGPR r -> (M=r, N=lane) lanes 0-15 ; (M=r+8, N=lane-16) lanes 16-31
    const int mAdd = (lane >> 4) * 8;
#pragma unroll
    for (int r = 0; r < 8; ++r)
        hp[(size_t)(tileM * 16 + r + mAdd) * 128 + wave * 16 + row] = c[r];
}

// ---------------------------------------------------------------------------
// Per-node attention coefficients: a_src[n,h] = sum_c hp[n,h,c]*att_src[h,c].
// One wave per node; flat layout of hp/att are both head-major so chunk j
// (features 32j..32j+31) == head j when H=4, or part of head 0 when H=1.
// ---------------------------------------------------------------------------
__global__ __launch_bounds__(256) void attn_coef(
    const float* __restrict__ hp, const float* __restrict__ att_s,
    const float* __restrict__ att_d, float* __restrict__ a_src,
    float* __restrict__ a_dst, int n, int heads)
{
    const int wid  = (int)((blockIdx.x * blockDim.x + threadIdx.x) >> 5);
    const int lane = threadIdx.x & 31;
    if (wid >= n) return;

    const float* hrow = hp + (size_t)wid * 128;
    float ss[4], sd[4];
#pragma unroll
    for (int j = 0; j < 4; ++j) {
        const float v = hrow[j * 32 + lane];
        ss[j] = v * att_s[j * 32 + lane];
        sd[j] = v * att_d[j * 32 + lane];
    }
#pragma unroll
    for (int j = 0; j < 4; ++j) {
#pragma unroll
        for (int m = 16; m >= 1; m >>= 1) {
            ss[j] += __shfl_xor(ss[j], m, 32);
            sd[j] += __shfl_xor(sd[j], m, 32);
        }
    }
    if (lane == 0) {
        if (heads == 4) {
#pragma unroll
            for (int j = 0; j < 4; ++j) {
                a_src[wid * 4 + j] = ss[j];
                a_dst[wid * 4 + j] = sd[j];
            }
        } else {
            a_src[wid] = ss[0] + ss[1] + ss[2] + ss[3];
            a_dst[wid] = sd[0] + sd[1] + sd[2] + sd[3];
        }
    }
}

// Order-preserving float<->uint map so segment-max can use integer atomicMax.
__device__ __forceinline__ unsigned f2ord(float f) {
    unsigned u = __float_as_uint(f);
    return (u & 0x80000000u) ? ~u : (u | 0x80000000u);
}
__device__ __forceinline__ float ord2f(unsigned u) {
    return (u & 0x80000000u) ? __uint_as_float(u & 0x7fffffffu)
                             : __uint_as_float(~u);
}

// ---------------------------------------------------------------------------
// Pass 1: e = leaky_relu(a_src[src]+a_dst[dst]); segment max over dst.
// One thread per (edge, head). Edges e >= E are self-loops (src=dst=e-E).
// ---------------------------------------------------------------------------
__global__ void edge_logits(
    const int* __restrict__ eidx, int E, int Etot,
    const float* __restrict__ a_src, const float* __restrict__ a_dst,
    float* __restrict__ ev, unsigned* __restrict__ emax, int heads)
{
    const int t = (int)(blockIdx.x * blockDim.x + threadIdx.x);
    if (t >= Etot * heads) return;
    const int e  = t / heads;
    const int hh = t - e * heads;
    const int s = (e < E) ? eidx[e]     : (e - E);
    const int d = (e < E) ? eidx[E + e] : (e - E);
    float x = a_src[s * heads + hh] + a_dst[d * heads + hh];
    x = (x > 0.f) ? x : NEG_SLOPE * x;
    ev[t] = x;
    atomicMax(&emax[d * heads + hh], f2ord(x));
}

// ---------------------------------------------------------------------------
// Pass 2: ex = exp(e - emax[dst]); segment sum (denominator).
// ---------------------------------------------------------------------------
__global__ void edge_exp(
    const int* __restrict__ eidx, int E, int Etot,
    float* __restrict__ ev, const unsigned* __restrict__ emax,
    float* __restrict__ denom, int heads)
{
    const int t = (int)(blockIdx.x * blockDim.x + threadIdx.x);
    if (t >= Etot * heads) return;
    const int e  = t / heads;
    const int hh = t - e * heads;
    const int d = (e < E) ? eidx[E + e] : (e - E);
    const float ex = expf(ev[t] - ord2f(emax[d * heads + hh]));
    ev[t] = ex;
    atomicAdd(&denom[d * heads + hh], ex);
}

// ---------------------------------------------------------------------------
// Pass 3: agg[dst] += hp[src] * alpha. One wave per edge; lane owns features
// {lane, lane+32, lane+64, lane+96}; alpha per head chunk.
// ---------------------------------------------------------------------------
__global__ __launch_bounds__(256) void edge_aggregate(
    const int* __restrict__ eidx, int E, int Etot,
    const float* __restrict__ hp, const float* __restrict__ ev,
    const float* __restrict__ denom, float* __restrict__ agg, int heads)
{
    const int wid  = (int)((blockIdx.x * blockDim.x + threadIdx.x) >> 5);
    const int lane = threadIdx.x & 31;
    if (wid >= Etot) return;
    const int e = wid;
    const int s = (e < E) ? eidx[e]     : (e - E);
    const int d = (e < E) ? eidx[E + e] : (e - E);
#pragma unroll
    for (int j = 0; j < 4; ++j) {
        const int   hj    = (heads == 4) ? j : 0;
        const float alpha = ev[e * heads + hj] / denom[d * heads + hj];
        const float v     = hp[(size_t)s * 128 + j * 32 + lane] * alpha;
        atomicAdd(&agg[(size_t)d * 128 + j * 32 + lane], v);
    }
}

// ---------------------------------------------------------------------------
// Finalize layers 0/1: bias, LayerNorm(128), ELU, +residual. One wave/node.
// Writes next-layer features (h_out) and the JK-concat output slice.
// ---------------------------------------------------------------------------
__global__ __launch_bounds__(256) void finalize_ln_elu_res(
    const float* __restrict__ agg, const float* __restrict__ bias,
    const float* __restrict__ ln_s, const float* __restrict__ ln_b,
    const float* __restrict__ res, float* __restrict__ h_out,
    float* __restrict__ out, int n, int out_off)
{
    const int wid  = (int)((blockIdx.x * blockDim.x + threadIdx.x) >> 5);
    const int lane = threadIdx.x & 31;
    if (wid >= n) return;

    float v[4], s = 0.f, s2 = 0.f;
#pragma unroll
    for (int j = 0; j < 4; ++j) {
        const float x = agg[(size_t)wid * 128 + j * 32 + lane] + bias[j * 32 + lane];
        v[j] = x; s += x; s2 += x * x;
    }
#pragma unroll
    for (int m = 16; m >= 1; m >>= 1) {
        s  += __shfl_xor(s,  m, 32);
        s2 += __shfl_xor(s2, m, 32);
    }
    const float mu  = s * (1.f / 128.f);
    const float var = s2 * (1.f / 128.f) - mu * mu;
    const float inv = rsqrtf(var + LN_EPS);
#pragma unroll
    for (int j = 0; j < 4; ++j) {
        const int i = j * 32 + lane;
        float x = (v[j] - mu) * inv * ln_s[i] + ln_b[i];
        x = (x > 0.f) ? x : (expf(x) - 1.f);             // ELU(alpha=1)
        x += res[(size_t)wid * 128 + i];                 // residual
        h_out[(size_t)wid * 128 + i]   = x;
        out[(size_t)wid * 384 + out_off + i] = x;
    }
}

// Finalize layer 2: bias add only, into the last concat slice.
__global__ void finalize_bias(
    const float* __restrict__ agg, const float* __restrict__ bias,
    float* __restrict__ out, int n, int out_off)
{
    const int t = (int)(blockIdx.x * blockDim.x + threadIdx.x);
    if (t >= n * 128) return;
    const int node = t >> 7, i = t & 127;
    out[(size_t)node * 384 + out_off + i] = agg[t] + bias[i];
}

// ---------------------------------------------------------------------------
extern "C" void kernel_launch(void* const* d_in, const int* in_sizes, int n_in,
                              void* d_out, int out_size, void* d_ws, size_t ws_size,
                              hipStream_t stream)
{
    (void)n_in; (void)out_size; (void)ws_size;

    const float* x    = (const float*)d_in[0];
    const int*   eidx = (const int*)d_in[1];
    const int n    = in_sizes[0] / 128;
    const int E    = in_sizes[1] / 2;
    const int Etot = E + n;

    const float* W[3]   = {(const float*)d_in[2], (const float*)d_in[8],  (const float*)d_in[14]};
    const float* as[3]  = {(const float*)d_in[3], (const float*)d_in[9],  (const float*)d_in[15]};
    const float* ad[3]  = {(const float*)d_in[4], (const float*)d_in[10], (const float*)d_in[16]};
    const float* bs[3]  = {(const float*)d_in[5], (const float*)d_in[11], (const float*)d_in[17]};
    const float* lns[2] = {(const float*)d_in[6], (const float*)d_in[12]};
    const float* lnb[2] = {(const float*)d_in[7], (const float*)d_in[13]};

    // Workspace layout (~56 MB)
    float* ws    = (float*)d_ws;
    float* h_cur = ws;               ws += (size_t)n * 128;
    float* hp    = ws;               ws += (size_t)n * 128;
    float* agg   = ws;               ws += (size_t)n * 128;
    float* a_src = ws;               ws += (size_t)n * 4;
    float* a_dst = ws;               ws += (size_t)n * 4;
    unsigned* emax = (unsigned*)ws;  ws += (size_t)n * 4;
    float* denom = ws;               ws += (size_t)n * 4;
    float* ev    = ws;               ws += (size_t)Etot * 4;

    float* out = (float*)d_out;

    for (int layer = 0; layer < 3; ++layer) {
        const int heads = (layer == 2) ? 1 : 4;
        const float* hin = (layer == 0) ? x : h_cur;

        gemm128_wmma<<<n / 16, 256, 0, stream>>>(hin, W[layer], hp);
        attn_coef<<<(n * 32 + 255) / 256, 256, 0, stream>>>(
            hp, as[layer], ad[layer], a_src, a_dst, n, heads);

        hipMemsetAsync(agg,   0, (size_t)n * 128 * sizeof(float),    stream);
        hipMemsetAsync(emax,  0, (size_t)n * heads * sizeof(unsigned), stream);
        hipMemsetAsync(denom, 0, (size_t)n * heads * sizeof(float),  stream);

        const int eth = Etot * heads;
        edge_logits<<<(eth + 255) / 256, 256, 0, stream>>>(
            eidx, E, Etot, a_src, a_dst, ev, emax, heads);
        edge_exp<<<(eth + 255) / 256, 256, 0, stream>>>(
            eidx, E, Etot, ev, emax, denom, heads);
        edge_aggregate<<<(Etot * 32 + 255) / 256, 256, 0, stream>>>(
            eidx, E, Etot, hp, ev, denom, agg, heads);

        if (layer < 2) {
            const float* res = (layer == 0) ? x : h_cur;
            finalize_ln_elu_res<<<(n * 32 + 255) / 256, 256, 0, stream>>>(
                agg, bs[layer], lns[layer], lnb[layer], res, h_cur, out, n, layer * 128);
        } else {
            finalize_bias<<<(n * 128 + 255) / 256, 256, 0, stream>>>(
                agg, bs[2], out, n, 256);
        }
    }
}